// LightGCN_76751065579690
// MI455X (gfx1250) — compile-verified
//
#include <hip/hip_runtime.h>
#include <math.h>

#define N_USER 100000
#define N_ITEM 100000
#define NNODE  200000
#define D      64
#define BATCH  4096
#define NPAIR  20480      // 4096 positives + 16384 negatives
#define REG_W  0.0001f
#define BPR_G  1e-10f

typedef __attribute__((ext_vector_type(2))) float v2f;
typedef __attribute__((ext_vector_type(8))) float v8f;

// ---------------------------------------------------------------------------
// Init: h = acc = concat(user_w, item_w); zero the reduction scalars.
// ---------------------------------------------------------------------------
__global__ void init_kernel(const float* __restrict__ uw, const float* __restrict__ iw,
                            float* __restrict__ h, float* __restrict__ acc,
                            float* __restrict__ scal) {
  long i = (long)blockIdx.x * blockDim.x + threadIdx.x;
  const long tot = (long)NNODE * D;
  if (i < tot) {
    float v = (i < (long)N_USER * D) ? uw[i] : iw[i - (long)N_USER * D];
    h[i]   = v;
    acc[i] = v;
  }
  if (i < 4) scal[i] = 0.0f;
}

__global__ void zero_kernel(float* __restrict__ p, long n) {
  long i = (long)blockIdx.x * blockDim.x + threadIdx.x;
  if (i < n) p[i] = 0.0f;
}

// ---------------------------------------------------------------------------
// Edge-parallel SpMM: one wave32 per edge. Gather h[col], scale by val,
// scatter-add into hn[row] with native f32 atomics (L2-resident: h+hn fit
// in the 192MB L2). Lane l handles dims l and l+32 -> coalesced 128B ops.
// ---------------------------------------------------------------------------
__global__ void edge_kernel(const float* __restrict__ h, float* __restrict__ hn,
                            const int* __restrict__ row, const int* __restrict__ col,
                            const float* __restrict__ val, int nnz) {
  int e    = (blockIdx.x * blockDim.x + threadIdx.x) >> 5;
  int lane = threadIdx.x & 31;
  if (e >= nnz) return;
  __builtin_prefetch(col + e + 2048, 0, 1);   // global_prefetch_b8 on edge stream
  __builtin_prefetch(val + e + 2048, 0, 1);
  int   r = row[e];
  int   c = col[e];
  float v = val[e];
  const float* src = h  + (long)c * D;
  float*       dst = hn + (long)r * D;
  unsafeAtomicAdd(dst + lane,      v * src[lane]);        // global_atomic_add_f32
  unsafeAtomicAdd(dst + lane + 32, v * src[lane + 32]);
}

// ---------------------------------------------------------------------------
// Row L2-normalize hn -> h, and acc += coef * normalized.  One wave per row,
// wave32 shuffle reduction for the squared norm.
// ---------------------------------------------------------------------------
__global__ void norm_acc_kernel(const float* __restrict__ hn, float* __restrict__ h,
                                float* __restrict__ acc, float coef) {
  int rowid = (blockIdx.x * blockDim.x + threadIdx.x) >> 5;
  int lane  = threadIdx.x & 31;
  if (rowid >= NNODE) return;
  const float* src = hn + (long)rowid * D;
  float a = src[lane];
  float b = src[lane + 32];
  float s = a * a + b * b;
  #pragma unroll
  for (int off = 16; off > 0; off >>= 1) s += __shfl_xor(s, off, 32);
  float inv = 1.0f / fmaxf(sqrtf(s), 1e-12f);
  float na = a * inv, nb = b * inv;
  long base = (long)rowid * D;
  h[base + lane]        = na;
  h[base + lane + 32]   = nb;
  acc[base + lane]      += coef * na;
  acc[base + lane + 32] += coef * nb;
}

// ---------------------------------------------------------------------------
// Pair scoring with V_WMMA_F32_16X16X4_F32.  Each wave owns 16 pairs:
// A rows = user embeddings, B cols = item embeddings, K looped over 64 in
// steps of 4.  The 16 scores are the diagonal of the 16x16 result tile,
// extracted via LDS using the documented C-layout
// (p<8: lane p / VGPR p ; p>=8: lane p+16 / VGPR p-8).
// NPAIR = 20480 = 1280 waves = 160 blocks of 256 -> no partial waves,
// EXEC is all-ones at every WMMA.
// ---------------------------------------------------------------------------
__global__ void __launch_bounds__(256)
score_kernel(const float* __restrict__ emb, const int* __restrict__ x,
             float* __restrict__ scores) {
  int lane = threadIdx.x & 31;
  int wloc = threadIdx.x >> 5;
  int wglb = blockIdx.x * 8 + wloc;
  int t0   = wglb * 16;

  // Pair id this lane supplies fragments for (lanes 16-31 mirror 0-15).
  int p  = t0 + (lane & 15);
  int in_pos = (p < BATCH) ? 1 : 0;
  int t2 = p - BATCH;
  int b  = in_pos ? p : (t2 >> 2);
  int s  = in_pos ? 0 : (1 + (t2 & 3));
  int u  = x[b * 24 + s * 4];          // x is (4096, 6, 4) int32: [u, i, 0, 0]
  int it = x[b * 24 + s * 4 + 1];
  const float* urow = emb + (long)u * D;
  const float* irow = emb + ((long)N_USER + it) * D;

  // f32 16x4 A layout: lanes 0-15 hold K=k0,k0+1; lanes 16-31 hold K=k0+2,k0+3.
  int kh = (lane >> 4) << 1;
  v8f c = {0.f, 0.f, 0.f, 0.f, 0.f, 0.f, 0.f, 0.f};
  #pragma unroll
  for (int k0 = 0; k0 < D; k0 += 4) {
    v2f a, bb;
    a.x  = urow[k0 + kh];
    a.y  = urow[k0 + kh + 1];
    bb.x = irow[k0 + kh];
    bb.y = irow[k0 + kh + 1];
    c = __builtin_amdgcn_wmma_f32_16x16x4_f32(
        /*neg_a=*/false, a, /*neg_b=*/false, bb,
        /*c_mod=*/(short)0, c, /*reuse_a=*/false, /*reuse_b=*/false);
  }

  __shared__ float tile[8][32][8];
  #pragma unroll
  for (int v = 0; v < 8; ++v) tile[wloc][lane][v] = c[v];
  __syncthreads();

  int pp = lane & 15;
  float sc = (pp < 8) ? tile[wloc][pp][pp] : tile[wloc][pp + 16][pp - 8];
  if (lane < 16) scores[t0 + pp] = sc;
}

// ---------------------------------------------------------------------------
// BPR loss terms + reduction into scal[2].
// ---------------------------------------------------------------------------
__global__ void loss_kernel(const float* __restrict__ scores, float* __restrict__ scal) {
  int b = blockIdx.x * blockDim.x + threadIdx.x;
  float term = 0.0f;
  if (b < BATCH) {
    float ps = scores[b];
    const float* ns = scores + BATCH + b * 4;
    float d   = ps - (ns[0] + ns[1] + ns[2] + ns[3]);
    float sig = 1.0f / (1.0f + expf(-d));
    term = -logf(BPR_G + sig);
  }
  #pragma unroll
  for (int off = 16; off > 0; off >>= 1) term += __shfl_xor(term, off, 32);
  if ((threadIdx.x & 31) == 0) unsafeAtomicAdd(&scal[2], term);
}

// ---------------------------------------------------------------------------
// Sum of squares (for Frobenius norms), grid-stride + wave reduction.
// ---------------------------------------------------------------------------
__global__ void sq_kernel(const float* __restrict__ w, long n, float* __restrict__ target) {
  long i      = (long)blockIdx.x * blockDim.x + threadIdx.x;
  long stride = (long)gridDim.x * blockDim.x;
  float s = 0.0f;
  for (; i < n; i += stride) { float v = w[i]; s += v * v; }
  #pragma unroll
  for (int off = 16; off > 0; off >>= 1) s += __shfl_xor(s, off, 32);
  if ((threadIdx.x & 31) == 0) unsafeAtomicAdd(target, s);
}

__global__ void final_kernel(const float* __restrict__ scal, float* __restrict__ out) {
  if (threadIdx.x == 0 && blockIdx.x == 0) {
    float emb = (sqrtf(scal[0]) + sqrtf(scal[1])) / (float)N_ITEM;
    out[0] = scal[2] / (float)BATCH + REG_W * emb;
  }
}

// ---------------------------------------------------------------------------
extern "C" void kernel_launch(void* const* d_in, const int* in_sizes, int n_in,
                              void* d_out, int out_size, void* d_ws, size_t ws_size,
                              hipStream_t stream) {
  const int*   x   = (const int*)d_in[0];
  const float* uw  = (const float*)d_in[1];
  const float* iw  = (const float*)d_in[2];
  const int*   row = (const int*)d_in[3];
  const int*   col = (const int*)d_in[4];
  const float* val = (const float*)d_in[5];
  int nnz = in_sizes[3];

  float* ws     = (float*)d_ws;
  const long tot = (long)NNODE * D;      // 12.8M floats
  float* h      = ws;
  float* hn     = h + tot;
  float* acc    = hn + tot;
  float* scores = acc + tot;
  float* scal   = scores + NPAIR;
  float* out    = (float*)d_out;

  const int TB = 256;
  init_kernel<<<(int)((tot + TB - 1) / TB), TB, 0, stream>>>(uw, iw, h, acc, scal);

  // out = out + h/(i+1) + h  -> coef = 1 + 1/(i+1)
  const float coefs[3] = {2.0f, 1.5f, 4.0f / 3.0f};
  for (int L = 0; L < 3; ++L) {
    zero_kernel<<<(int)((tot + TB - 1) / TB), TB, 0, stream>>>(hn, tot);
    int eblocks = (nnz + 7) / 8;                        // 8 edges (waves) / block
    edge_kernel<<<eblocks, TB, 0, stream>>>(h, hn, row, col, val, nnz);
    norm_acc_kernel<<<(NNODE + 7) / 8, TB, 0, stream>>>(hn, h, acc, coefs[L]);
  }

  score_kernel<<<NPAIR / 16 / 8, TB, 0, stream>>>(acc, x, scores);   // 160 blocks
  loss_kernel<<<(BATCH + TB - 1) / TB, TB, 0, stream>>>(scores, scal);
  sq_kernel<<<512, TB, 0, stream>>>(uw, (long)N_USER * D, &scal[0]);
  sq_kernel<<<512, TB, 0, stream>>>(iw, (long)N_ITEM * D, &scal[1]);
  final_kernel<<<1, 1, 0, stream>>>(scal, out);
}